// ConnectivityLoss_4947802325235
// MI455X (gfx1250) — compile-verified
//
#include <hip/hip_runtime.h>

// ConnectivityLoss — fully fused soft-skeletonize + connectivity MSE for MI455X (gfx1250).
// Stencil workload => no WMMA (no matmul structure). CDNA5 path used: async
// global->LDS DMA (global_load_async_to_lds_b32 / ASYNCcnt / s_wait_asynccnt).

#define N_IMG 32
#define H 512
#define W 512
#define TILE 64
#define TPB 256
#define R 11                     // halo: 5 iters * 2 + 1 (sum-pool)
#define AW (TILE + 2 * R)        // 86
#define AST 88                   // padded LDS row stride (floats)
#define PIX_PER_THREAD ((TILE * TILE) / TPB) // 16
#define BIGV 1.0e30f

// LDS byte offset of a generic pointer to __shared__ data: on AMDGPU the
// generic shared aperture is {SHARED_BASE[15:0],16'b0} in the high 32 bits,
// so the low 32 bits ARE the LDS byte offset.
__device__ __forceinline__ unsigned lds_off32(const void* p) {
    return (unsigned)(unsigned long long)p;
}

// CDNA5 async DMA: one dword from global memory into LDS, tracked by ASYNCcnt.
__device__ __forceinline__ void async_load_f32(void* lds_dst, const float* gsrc) {
    unsigned l = lds_off32(lds_dst);
    unsigned long long g = (unsigned long long)gsrc;
    asm volatile("global_load_async_to_lds_b32 %0, %1, off"
                 :
                 : "v"(l), "v"(g)
                 : "memory");
}

__device__ __forceinline__ void wait_async0() {
    asm volatile("s_wait_asynccnt 0" ::: "memory");
}

__global__ __launch_bounds__(TPB)
void skel_tile_kernel(const float* __restrict__ pred,
                      const float* __restrict__ tgt,
                      float* __restrict__ partials) {
    __shared__ float A[AW * AST];   // x / skeleton, halo 11 (+BIG outside image)
    __shared__ float Bm[AW * AST];  // 3x3 min-pool buffer
    __shared__ float red[TPB];

    const int tid  = threadIdx.x;
    const int img  = blockIdx.y;
    const int tile = blockIdx.x;
    const int ty0  = (tile >> 3) * TILE;
    const int tx0  = (tile & 7) * TILE;
    const size_t imgOff = (size_t)img * H * W;

    float skP[PIX_PER_THREAD];  // pred skeleton per owned pixel
    float ctP[PIX_PER_THREAD];  // pred neighbor_count per owned pixel
    float accS = 0.f, accE = 0.f, accC = 0.f;

    for (int pass = 0; pass < 2; ++pass) {
        const float* src = (pass == 0 ? pred : tgt) + imgOff;

        // ---- stage tile + halo into LDS via async DMA; image-OOB cells = +BIG ----
        for (int idx = tid; idx < AW * AW; idx += TPB) {
            int ly = idx / AW;
            int lx = idx - ly * AW;
            int gy = ty0 + ly - R;
            int gx = tx0 + lx - R;
            float* dst = &A[ly * AST + lx];
            if ((unsigned)gy < (unsigned)H && (unsigned)gx < (unsigned)W) {
                async_load_f32(dst, src + (size_t)gy * W + gx);
            } else {
                *dst = BIGV;
            }
        }
        wait_async0();
        __syncthreads();

        // ---- 5 soft-skeleton iterations ----
        for (int it = 0; it < 5; ++it) {
            // 3x3 min-pool -> Bm over halo-10 region (84x84). OOB cells -> 0.
            for (int idx = tid; idx < 84 * 84; idx += TPB) {
                int yq = idx / 84;
                int y  = yq + 1;
                int x  = idx - yq * 84 + 1;
                int gy = ty0 + y - R;
                int gx = tx0 + x - R;
                float v = 0.f;
                if ((unsigned)gy < (unsigned)H && (unsigned)gx < (unsigned)W) {
                    const float* c = &A[y * AST + x];
                    float m0 = fminf(fminf(c[-AST - 1], c[-AST]), c[-AST + 1]);
                    float m1 = fminf(fminf(c[-1],       c[0]),    c[1]);
                    float m2 = fminf(fminf(c[AST - 1],  c[AST]),  c[AST + 1]);
                    v = fminf(fminf(m0, m1), m2);
                }
                Bm[y * AST + x] = v;
            }
            __syncthreads();

            // contour = relu(maxpool3(Bm) - Bm); x = relu(x - contour), halo-9 (82x82)
            for (int idx = tid; idx < 82 * 82; idx += TPB) {
                int yq = idx / 82;
                int y  = yq + 2;
                int x  = idx - yq * 82 + 2;
                int gy = ty0 + y - R;
                int gx = tx0 + x - R;
                if ((unsigned)gy < (unsigned)H && (unsigned)gx < (unsigned)W) {
                    const float* c = &Bm[y * AST + x];
                    float m0 = fmaxf(fmaxf(c[-AST - 1], c[-AST]), c[-AST + 1]);
                    float m1 = fmaxf(fmaxf(c[-1],       c[0]),    c[1]);
                    float m2 = fmaxf(fmaxf(c[AST - 1],  c[AST]),  c[AST + 1]);
                    float mx = fmaxf(fmaxf(m0, m1), m2);
                    float contour = fmaxf(mx - c[0], 0.f);
                    float nx = A[y * AST + x] - contour;
                    A[y * AST + x] = fmaxf(nx, 0.f);
                }
            }
            __syncthreads();
        }

        // ---- zero image-OOB ring in halo-1 region so sum-pool pads with 0 ----
        for (int idx = tid; idx < 66 * 66; idx += TPB) {
            int yq = idx / 66;
            int y  = yq + 10;
            int x  = idx - yq * 66 + 10;
            int gy = ty0 + y - R;
            int gx = tx0 + x - R;
            if (!((unsigned)gy < (unsigned)H && (unsigned)gx < (unsigned)W))
                A[y * AST + x] = 0.f;
        }
        __syncthreads();

        // ---- 3x3 sum-pool + classification on owned pixels ----
        for (int k = 0; k < PIX_PER_THREAD; ++k) {
            int p   = tid + k * TPB;
            int row = p >> 6;
            int col = p & 63;
            const float* c = &A[(row + R) * AST + (col + R)];
            float cnt = c[-AST - 1] + c[-AST] + c[-AST + 1]
                      + c[-1]       + c[0]    + c[1]
                      + c[AST - 1]  + c[AST]  + c[AST + 1];
            float skel = c[0];
            if (pass == 0) {
                skP[k] = skel;
                ctP[k] = cnt;
            } else {
                float onT = (skel   > 0.5f) ? 1.f : 0.f;
                float onP = (skP[k] > 0.5f) ? 1.f : 0.f;
                float eT = (cnt    == 2.0f) ? onT : 0.f;
                float eP = (ctP[k] == 2.0f) ? onP : 0.f;
                float cT = (cnt    >= 4.0f) ? onT : 0.f;
                float cP = (ctP[k] >= 4.0f) ? onP : 0.f;
                float d0 = skel - skP[k];
                float d1 = eT - eP;
                float d2 = cT - cP;
                accS += d0 * d0;
                accE += d1 * d1;
                accC += d2 * d2;
            }
        }
        __syncthreads();  // A/Bm reused by next pass
    }

    // ---- deterministic block reduction, one partial-triple per block ----
    float acc3[3] = {accS, accE, accC};
    const int bid = blockIdx.y * gridDim.x + blockIdx.x;
    for (int j = 0; j < 3; ++j) {
        red[tid] = acc3[j];
        __syncthreads();
        for (int st = TPB / 2; st > 0; st >>= 1) {
            if (tid < st) red[tid] += red[tid + st];
            __syncthreads();
        }
        if (tid == 0) partials[bid * 3 + j] = red[0];
        __syncthreads();
    }
}

__global__ __launch_bounds__(TPB)
void final_reduce_kernel(const float* __restrict__ partials, int nblk,
                         float* __restrict__ out) {
    __shared__ float red[TPB * 3];
    float a0 = 0.f, a1 = 0.f, a2 = 0.f;
    for (int b = threadIdx.x; b < nblk; b += TPB) {
        a0 += partials[b * 3 + 0];
        a1 += partials[b * 3 + 1];
        a2 += partials[b * 3 + 2];
    }
    red[threadIdx.x * 3 + 0] = a0;
    red[threadIdx.x * 3 + 1] = a1;
    red[threadIdx.x * 3 + 2] = a2;
    __syncthreads();
    for (int st = TPB / 2; st > 0; st >>= 1) {
        if (threadIdx.x < st) {
            red[threadIdx.x * 3 + 0] += red[(threadIdx.x + st) * 3 + 0];
            red[threadIdx.x * 3 + 1] += red[(threadIdx.x + st) * 3 + 1];
            red[threadIdx.x * 3 + 2] += red[(threadIdx.x + st) * 3 + 2];
        }
        __syncthreads();
    }
    if (threadIdx.x == 0) {
        const float inv = 1.0f / (float)((long long)N_IMG * H * W);
        out[0] = (0.6f * red[0] + 0.2f * red[1] + 0.2f * red[2]) * inv;
    }
}

extern "C" void kernel_launch(void* const* d_in, const int* in_sizes, int n_in,
                              void* d_out, int out_size, void* d_ws, size_t ws_size,
                              hipStream_t stream) {
    (void)in_sizes; (void)n_in; (void)out_size; (void)ws_size;
    const float* pred = (const float*)d_in[0];
    const float* tgt  = (const float*)d_in[1];
    float* partials   = (float*)d_ws;           // 2048 blocks * 3 floats = 24 KB

    dim3 grid((W / TILE) * (H / TILE) / 1, N_IMG);  // 64 tiles/image x 32 images
    grid.x = (W / TILE) * (H / TILE);               // 8*8 = 64
    skel_tile_kernel<<<grid, TPB, 0, stream>>>(pred, tgt, partials);
    final_reduce_kernel<<<1, TPB, 0, stream>>>(partials, (int)grid.x * N_IMG,
                                               (float*)d_out);
}